// BoundSigmoid_9990093930707
// MI455X (gfx1250) — compile-verified
//
#include <hip/hip_runtime.h>
#include <cstdint>

#define B_    8
#define S_    32
#define CHW   150528      /* 3*224*224 */
#define NA    38535168    /* B*S*CHW  */
#define TILE  1024
#define DEPTH 4
#define EPSF  1e-12f

typedef float v2f __attribute__((ext_vector_type(2)));
typedef float v8f __attribute__((ext_vector_type(8)));

__device__ __forceinline__ float sigf(float x) {
    return 1.0f / (1.0f + expf(-x));
}

extern "C" __global__ void zero_bias_kernel(float* __restrict__ out) {
    const int t = threadIdx.x;
    if (t < 256) {
        out[(size_t)NA + t] = 0.0f;                 // lbias [B,S]
        out[2 * (size_t)NA + 256 + t] = 0.0f;       // ubias [B,S]
    }
}

extern "C" __global__ __launch_bounds__(256)
void bound_sigmoid_fused(const float* __restrict__ lastlA,
                         const float* __restrict__ lastuA,
                         const float* __restrict__ xlow,
                         const float* __restrict__ xdelta,
                         const float* __restrict__ dlo,
                         const float* __restrict__ dhi,
                         float* __restrict__ out)
{
    const int t    = threadIdx.x;
    const int b    = blockIdx.y;
    const int base = blockIdx.x * TILE;

    __shared__ float sTabL[1008];
    __shared__ float sTabU[1008];
    __shared__ __align__(16) float sTile[DEPTH][2][TILE];  // async-load ring buffer

    for (int i = t; i < 1005; i += 256) { sTabL[i] = dlo[i]; sTabU[i] = dhi[i]; }
    __syncthreads();

    // ---- per-thread relaxation coefficients for its 4 spatial elements (reused over all S) ----
    const int e0 = base + t * 4;
    const float4 xl4 = *(const float4*)(xlow   + (size_t)b * CHW + e0);
    const float4 xd4 = *(const float4*)(xdelta + (size_t)b * CHW + e0);

    float LW[4], LB[4], UW[4], UB[4];
    {
        float lo[4] = {xl4.x, xl4.y, xl4.z, xl4.w};
        float de[4] = {xd4.x, xd4.y, xd4.z, xd4.w};
#pragma unroll
        for (int c = 0; c < 4; ++c) {
            float lower = lo[c];
            float upper = lower + de[c];
            float yl = sigf(lower);
            float yu = sigf(upper);
            float m  = 0.5f * (lower + upper);
            float ym = sigf(m);
            float km = ym * (1.0f - ym);
            float kc = (yu - yl) / (upper - lower + EPSF);
            float lw, lb, uw, ub;
            if (lower > 0.0f) {            // concave side: chord below, tangent above
                lw = kc; lb = fmaf(-lower, kc, yl);
                uw = km; ub = fmaf(-m, km, ym);
            } else if (upper < 0.0f) {     // convex side: tangent below, chord above
                lw = km; lb = fmaf(-m, km, ym);
                uw = kc; ub = fmaf(-lower, kc, yl);
            } else {                        // cross-zero: precomputed tangent points
                int iu = (int)(upper / 0.01f);
                iu = iu < 0 ? 0 : (iu > 1004 ? 1004 : iu);
                float d  = sTabL[iu];
                float yd = sigf(d);
                float k  = (yd - yl) / (d - lower + EPSF);
                lw = k; lb = fmaf(-d, k, yd);
                int il = (int)(lower / -0.01f);
                il = il < 0 ? 0 : (il > 1004 ? 1004 : il);
                d  = sTabU[il];
                yd = sigf(d);
                k  = (yd - yu) / (d - upper + EPSF);
                uw = k; ub = fmaf(-d, k, yd);
            }
            LW[c] = lw; LB[c] = lb; UW[c] = uw; UB[c] = ub;
        }
    }

    // ---- async-to-LDS streaming pipeline over the S dimension ----
    const uint32_t ldsRow   = (uint32_t)(uintptr_t)(&sTile[0][0][0]) + (uint32_t)t * 16u;
    const uint64_t gL       = (uint64_t)(uintptr_t)lastlA;
    const uint64_t gU       = (uint64_t)(uintptr_t)lastuA;
    const uint32_t rowBytes = (uint32_t)CHW * 4u;
    const uint32_t goff0    = ((uint32_t)b * (uint32_t)S_ * (uint32_t)CHW + (uint32_t)e0) * 4u;

#define ISSUE_PAIR(p, sidx) do {                                                        \
        uint32_t go = goff0 + (uint32_t)(sidx) * rowBytes;                              \
        uint32_t dl = ldsRow + (uint32_t)(p) * (2u * TILE * 4u);                        \
        uint32_t du = dl + (uint32_t)(TILE * 4u);                                       \
        asm volatile("global_load_async_to_lds_b128 %0, %1, %2"                         \
                     :: "v"(dl), "v"(go), "s"(gL) : "memory");                          \
        asm volatile("global_load_async_to_lds_b128 %0, %1, %2"                         \
                     :: "v"(du), "v"(go), "s"(gU) : "memory");                          \
    } while (0)

#pragma unroll
    for (int p = 0; p < DEPTH; ++p) ISSUE_PAIR(p, p);

    float* outLA = out;
    float* outLB = out + (size_t)NA;                  // lbias [B,S]
    float* outUA = out + (size_t)NA + 256;
    float* outUB = out + 2 * (size_t)NA + 256;        // ubias [B,S]

    const int lane = t & 31;

    // WMMA reduction selector: B[k][0] = 1 for all k. Column N=0 of each B VGPR lives
    // in lanes 0 and 16 (rows stripe N across 16-lane halves), so this is independent
    // of which K rows map to which VGPR half.
    const float bsel = (lane == 0 || lane == 16) ? 1.0f : 0.0f;
    v2f bmat; bmat.x = bsel; bmat.y = bsel;
    const v8f zc = {};

    for (int s = 0; s < S_; ++s) {
        const int buf = s & (DEPTH - 1);
        // wait until the pair for iteration s has landed in LDS (per-wave, in-order)
        if (s <= S_ - DEPTH) asm volatile("s_wait_asynccnt 6" ::: "memory");
        else                 asm volatile("s_wait_asynccnt 0" ::: "memory");

        const float4 la4 = *(const float4*)(&sTile[buf][0][t * 4]);
        const float4 ua4 = *(const float4*)(&sTile[buf][1][t * 4]);
        float la[4] = {la4.x, la4.y, la4.z, la4.w};
        float ua[4] = {ua4.x, ua4.y, ua4.z, ua4.w};

        float ol[4], ou[4];
        float accL = 0.0f, accU = 0.0f;
#pragma unroll
        for (int c = 0; c < 4; ++c) {
            const bool pl = la[c] > 0.0f;
            float w  = pl ? LW[c] : UW[c];
            float bb = pl ? LB[c] : UB[c];
            ol[c] = la[c] * w;
            accL  = fmaf(la[c], bb, accL);
            const bool pu = ua[c] > 0.0f;
            w  = pu ? UW[c] : LW[c];
            bb = pu ? UB[c] : LB[c];
            ou[c] = ua[c] * w;
            accU  = fmaf(ua[c], bb, accU);
        }

        const size_t gidx = (size_t)(b * S_ + s) * CHW + e0;
        *(float4*)(outLA + gidx) = make_float4(ol[0], ol[1], ol[2], ol[3]);
        *(float4*)(outUA + gidx) = make_float4(ou[0], ou[1], ou[2], ou[3]);

        // refill the ring slot we just consumed (close WAR on LDS first)
        if (s + DEPTH < S_) {
            asm volatile("s_wait_dscnt 0" ::: "memory");
            ISSUE_PAIR(buf, s + DEPTH);
        }

        // ---- wave32 bias reduction via V_WMMA_F32_16X16X4_F32 ----
        // A.x = per-lane partial, A.y = 0:
        //   D[m][0] = acc[m] + acc[m+16]  (m = 0..15)
        // Summing the 8 D VGPRs per lane leaves the m=0..7 total in lane 0 and the
        // m=8..15 total in lane 16; one xor-16 shuffle combines them. EXEC is all-1s here.
        v2f aL; aL.x = accL; aL.y = 0.0f;
        v2f aU; aU.x = accU; aU.y = 0.0f;
        v8f dL = __builtin_amdgcn_wmma_f32_16x16x4_f32(false, aL, false, bmat,
                                                       (short)0, zc, false, false);
        v8f dU = __builtin_amdgcn_wmma_f32_16x16x4_f32(false, aU, false, bmat,
                                                       (short)0, zc, false, false);
        float rL = ((dL[0] + dL[1]) + (dL[2] + dL[3])) + ((dL[4] + dL[5]) + (dL[6] + dL[7]));
        float rU = ((dU[0] + dU[1]) + (dU[2] + dU[3])) + ((dU[4] + dU[5]) + (dU[6] + dU[7]));
        rL += __shfl_xor(rL, 16, 32);
        rU += __shfl_xor(rU, 16, 32);
        if (lane == 0) {
            atomicAdd(outLB + b * S_ + s, rL);
            atomicAdd(outUB + b * S_ + s, rU);
        }
    }
#undef ISSUE_PAIR
}

extern "C" void kernel_launch(void* const* d_in, const int* in_sizes, int n_in,
                              void* d_out, int out_size, void* d_ws, size_t ws_size,
                              hipStream_t stream) {
    (void)in_sizes; (void)n_in; (void)out_size; (void)d_ws; (void)ws_size;
    const float* lastlA = (const float*)d_in[0];
    const float* lastuA = (const float*)d_in[1];
    const float* xlow   = (const float*)d_in[2];
    const float* xdelta = (const float*)d_in[3];
    const float* dlo    = (const float*)d_in[4];
    const float* dhi    = (const float*)d_in[5];
    float* out = (float*)d_out;

    hipLaunchKernelGGL(zero_bias_kernel, dim3(1), dim3(256), 0, stream, out);
    dim3 grid(CHW / TILE, B_);
    hipLaunchKernelGGL(bound_sigmoid_fused, grid, dim3(256), 0, stream,
                       lastlA, lastuA, xlow, xdelta, dlo, dhi, out);
}